// Qwen3NextGatedDeltaNet_54494545052255
// MI455X (gfx1250) — compile-verified
//
#include <hip/hip_runtime.h>

#define B_   128
#define HK_  16
#define HV_  32
#define D_   128
#define KW_  4
#define QKV_ (D_*HK_*2 + D_*HV_)   // 8192

typedef float v2f __attribute__((ext_vector_type(2)));
typedef float v8f __attribute__((ext_vector_type(8)));

// ---------------------------------------------------------------------------
// Kernel 1: 4-tap conv + SiLU, then L2-normalize q/k heads.
// grid = B*64 blocks, 128 threads (one (b, qkv-head) per block).
// ---------------------------------------------------------------------------
__global__ __launch_bounds__(128)
void conv_norm_kernel(const float* __restrict__ mixed_qkv,
                      const float* __restrict__ conv_state,
                      const float* __restrict__ conv_w,
                      float* __restrict__ ws_q,
                      float* __restrict__ ws_k,
                      float* __restrict__ ws_v) {
  int blk  = blockIdx.x;
  int b    = blk >> 6;      // / 64
  int head = blk & 63;
  int d    = threadIdx.x;   // 0..127
  int c    = head * D_ + d;

  const float* cs = conv_state + (size_t)b * (KW_ - 1) * QKV_ + c;
  float s = cs[0]        * conv_w[c * KW_ + 0]
          + cs[QKV_]     * conv_w[c * KW_ + 1]
          + cs[2 * QKV_] * conv_w[c * KW_ + 2]
          + mixed_qkv[(size_t)b * QKV_ + c] * conv_w[c * KW_ + 3];
  float x = s / (1.0f + __expf(-s));   // silu

  __shared__ float red[D_];
  if (head < 2 * HK_) {                 // q or k head: l2norm over D
    red[d] = x * x;
    __syncthreads();
    #pragma unroll
    for (int st = 64; st > 0; st >>= 1) {
      if (d < st) red[d] += red[d + st];
      __syncthreads();
    }
    float y = x * rsqrtf(red[0] + 1e-6f);
    if (head < HK_) ws_q[((size_t)b * HK_ + head) * D_ + d] = y;
    else            ws_k[((size_t)b * HK_ + (head - HK_)) * D_ + d] = y;
  } else {
    ws_v[((size_t)b * HV_ + (head - 2 * HK_)) * D_ + d] = x;
  }
}

// ---------------------------------------------------------------------------
// Kernel 1b: per-(b,h) gate scalars.
// ---------------------------------------------------------------------------
__global__ void gate_kernel(const float* __restrict__ a_in,
                            const float* __restrict__ b_in,
                            const float* __restrict__ dt_bias,
                            const float* __restrict__ alog,
                            float* __restrict__ ws_eg,
                            float* __restrict__ ws_beta) {
  int i = blockIdx.x * blockDim.x + threadIdx.x;
  if (i >= B_ * HV_) return;
  int h = i & (HV_ - 1);
  float av = a_in[i] + dt_bias[h];
  float sp = (av > 20.0f) ? av : log1pf(__expf(av));   // softplus
  float g  = -__expf(alog[h]) * sp;
  ws_eg[i]   = __expf(g);
  ws_beta[i] = 1.0f / (1.0f + __expf(-b_in[i]));
}

// ---------------------------------------------------------------------------
// Kernel 2: per (b,h) — stream S (128x128 fp32) once, compute
//   y_q = S@qe, y_k = S@ke   via v_wmma_f32_16x16x4_f32 (cols 0/1 of B),
// then out = (expg*y_q + beta*(v - expg*y_k) * (ke.qe)) * D^-0.5.
// grid = B*HV blocks, 256 threads = 8 waves; wave w owns rows 16w..16w+15.
// ---------------------------------------------------------------------------
__global__ __launch_bounds__(256)
void delta_step_kernel(const float* __restrict__ S,
                       const float* __restrict__ ws_q,
                       const float* __restrict__ ws_k,
                       const float* __restrict__ ws_v,
                       const float* __restrict__ ws_eg,
                       const float* __restrict__ ws_beta,
                       float* __restrict__ out) {
  int bh   = blockIdx.x;      // b*HV + h
  int b    = bh >> 5;
  int h    = bh & 31;
  int tid  = threadIdx.x;
  int lane = tid & 31;
  int wave = tid >> 5;

  __shared__ __align__(16) float bmat[3][D_];  // row0=qe, row1=ke, row2=zeros
  __shared__ float yq[D_];
  __shared__ float yk[D_];
  __shared__ float dred[D_];
  __shared__ float sdot;

  int h2 = h >> 1;                              // repeat(q,2) / repeat(k,2)
  if (tid < D_) {
    bmat[0][tid] = ws_q[((size_t)b * HK_ + h2) * D_ + tid];
    bmat[2][tid] = 0.0f;
  } else {
    bmat[1][tid - D_] = ws_k[((size_t)b * HK_ + h2) * D_ + (tid - D_)];
  }
  __syncthreads();

  // dot_qk = ke . qe (rank-1 update collapses to this scalar)
  if (tid < D_) dred[tid] = bmat[0][tid] * bmat[1][tid];
  __syncthreads();
  #pragma unroll
  for (int st = 64; st > 0; st >>= 1) {
    if (tid < st) dred[tid] += dred[tid + st];
    __syncthreads();
  }
  if (tid == 0) sdot = dred[0];

  // WMMA fragment addressing (16x16x4 f32):
  //   A: lane = row M (0..15), vgpr pair = K=kbase+2*kk, kbase+2*kk+1
  //   B: lane = col N (0..15), same K striping; cols >=2 read the zero row
  int n     = lane & 15;
  int kk    = lane >> 4;
  int rbase = wave * 16;
  const float* arow = S + (((size_t)bh) * D_ + (rbase + n)) * D_ + 2 * kk;
  const float* brow = &bmat[n < 2 ? n : 2][2 * kk];

  v8f acc = {0.f, 0.f, 0.f, 0.f, 0.f, 0.f, 0.f, 0.f};
  #pragma unroll 4
  for (int it = 0; it < 32; ++it) {
    v2f av = __builtin_nontemporal_load((const v2f*)(arow + 4 * it));  // stream-once
    v2f bv = *(const v2f*)(brow + 4 * it);
    acc = __builtin_amdgcn_wmma_f32_16x16x4_f32(
        /*neg_a=*/false, av, /*neg_b=*/false, bv,
        /*c_mod=*/(short)0, acc, /*reuse_a=*/false, /*reuse_b=*/false);
  }

  // C layout: lanes 0-15 = col N, vgpr r -> M=r; lanes 16-31 -> M=r+8.
  if (n == 0) {
    #pragma unroll
    for (int r = 0; r < 8; ++r) yq[rbase + 8 * kk + r] = acc[r];
  } else if (n == 1) {
    #pragma unroll
    for (int r = 0; r < 8; ++r) yk[rbase + 8 * kk + r] = acc[r];
  }
  __syncthreads();

  if (tid < D_) {
    float eg    = ws_eg[bh];
    float bt    = ws_beta[bh];
    float vin   = ws_v[(size_t)bh * D_ + tid];
    float yqv   = eg * yq[tid];
    float ykv   = eg * yk[tid];
    float delta = (vin - ykv) * bt;
    out[(size_t)bh * D_ + tid] = (yqv + delta * sdot) * 0.08838834764831845f; // D^-0.5
  }
}

// ---------------------------------------------------------------------------
extern "C" void kernel_launch(void* const* d_in, const int* in_sizes, int n_in,
                              void* d_out, int out_size, void* d_ws, size_t ws_size,
                              hipStream_t stream) {
  const float* mixed_qkv  = (const float*)d_in[0];
  const float* b_in       = (const float*)d_in[1];
  const float* a_in       = (const float*)d_in[2];
  const float* conv_state = (const float*)d_in[3];
  const float* ssm_state  = (const float*)d_in[4];
  const float* conv_w     = (const float*)d_in[5];
  const float* dt_bias    = (const float*)d_in[6];
  const float* alog       = (const float*)d_in[7];
  float* out = (float*)d_out;

  float* ws = (float*)d_ws;
  size_t off = 0;
  float* ws_q    = ws + off; off += (size_t)B_ * HK_ * D_;
  float* ws_k    = ws + off; off += (size_t)B_ * HK_ * D_;
  float* ws_v    = ws + off; off += (size_t)B_ * HV_ * D_;
  float* ws_eg   = ws + off; off += (size_t)B_ * HV_;
  float* ws_beta = ws + off; off += (size_t)B_ * HV_;

  conv_norm_kernel<<<B_ * (2 * HK_ + HV_), D_, 0, stream>>>(
      mixed_qkv, conv_state, conv_w, ws_q, ws_k, ws_v);
  gate_kernel<<<(B_ * HV_ + 255) / 256, 256, 0, stream>>>(
      a_in, b_in, dt_bias, alog, ws_eg, ws_beta);
  delta_step_kernel<<<B_ * HV_, 256, 0, stream>>>(
      ssm_state, ws_q, ws_k, ws_v, ws_eg, ws_beta, out);
}